// VADetector_75110388072636
// MI455X (gfx1250) — compile-verified
//
#include <hip/hip_runtime.h>
#include <hip/hip_bf16.h>

// Problem constants (from reference): N_STATES=256, MEMORY_LENGTH=8, SNR_DB=10
// sigma2 = 10^-(10/10) = 0.1  ->  1/(2*sigma2) = 5.0
#define INV2SIG2 5.0f

typedef float v8f __attribute__((ext_vector_type(8)));
typedef float v2f __attribute__((ext_vector_type(2)));

// prior[s] = sum_m (1-2*bit(s,m)) * h[m], bits MSB-first
__device__ __forceinline__ float state_prior(int s, const float hh[8]) {
    float p = 0.0f;
#pragma unroll
    for (int m = 0; m < 8; ++m) {
        float sym = ((s >> (7 - m)) & 1) ? -1.0f : 1.0f;
        p = fmaf(sym, hh[m], p);
    }
    return p;
}

// ---------------------------------------------------------------------------
// Parallel likelihood pass via V_WMMA_F32_16X16X4_F32.
// One wave handles a 16-t x 256-s slab: A[t,:]=[a*rx^2, rx, 1, 0],
// B[:,s]=[1, -2a*p, a*p^2, 0]  =>  D[t,s] = a*(rx-p)^2 = cost.
// The 16 s-tiles are processed in a NON-unrolled loop so only the running
// sum(exp(-cost)) / max / argmax accumulators stay live (no spills).
// prior(s) is split into a uniform part (tile nibble, SALU-float) and a
// loop-invariant lane part.
// ---------------------------------------------------------------------------
__global__ void va_conf_kernel(const float* __restrict__ rx,
                               const float* __restrict__ h,
                               float* __restrict__ out_cb,
                               float* __restrict__ out_cw, int T) {
    float hh[8];
#pragma unroll
    for (int m = 0; m < 8; ++m) hh[m] = h[m];

    const int lane  = threadIdx.x & 31;
    const int wave  = threadIdx.x >> 5;
    const int tbase = (blockIdx.x * 8 + wave) * 16;   // uniform per wave
    if (tbase >= T) return;                           // wave-uniform exit

    const int M = lane & 15;
    float rv = 0.0f;
    if (tbase + M < T) rv = rx[tbase + M];

    // A matrix (16x4 f32): VGPR0 = K0 (lanes 0-15) / K2 (lanes 16-31),
    //                      VGPR1 = K1 (lanes 0-15) / K3 (lanes 16-31)
    v2f A;
    if (lane < 16) { A.x = INV2SIG2 * rv * rv; A.y = rv; }
    else           { A.x = 1.0f;               A.y = 0.0f; }

    // Loop-invariant lane part of the prior: bits 3..0 of s (taps h[4..7]).
    const int L = lane & 15;
    float p_lo = 0.0f;
#pragma unroll
    for (int m = 4; m < 8; ++m)
        p_lo += ((L >> (7 - m)) & 1) ? -hh[m] : hh[m];

    // Running reductions over the 16 s-tiles (tile-major order => strict '>'
    // keeps the first occurrence = smallest state, matching jnp.argmax).
    float sum[8], mx[8]; int si[8];
#pragma unroll
    for (int k = 0; k < 8; ++k) { sum[k] = 0.0f; mx[k] = -1.0f; si[k] = 0; }

#pragma unroll 1   // real loop: keeps live set small, prevents spill-inducing
    for (int st = 0; st < 16; ++st) {
        // Uniform part of the prior: bits 7..4 of s == st (taps h[0..3]).
        float p_hi = 0.0f;
#pragma unroll
        for (int m = 0; m < 4; ++m)
            p_hi += ((st >> (3 - m)) & 1) ? -hh[m] : hh[m];
        const float p = p_hi + p_lo;

        v2f B;
        if (lane < 16) { B.x = 1.0f;             B.y = -2.0f * INV2SIG2 * p; }
        else           { B.x = INV2SIG2 * p * p; B.y = 0.0f; }
        v8f c = {};
        v8f d = __builtin_amdgcn_wmma_f32_16x16x4_f32(
            false, A, false, B, (short)0, c, false, false);
#pragma unroll
        for (int k = 0; k < 8; ++k) {
            const float e = __expf(-d[k]);        // reference's unnormalized p
            sum[k] += e;
            if (e > mx[k]) { mx[k] = e; si[k] = st * 16 + L; }
        }
    }

    // Cross-lane reduction within each 16-lane half (xor masks 1..8 keep the
    // two M-halves of the C layout separate); argmax ties -> smaller state.
#pragma unroll
    for (int off = 1; off < 16; off <<= 1) {
#pragma unroll
        for (int k = 0; k < 8; ++k) {
            sum[k] += __shfl_xor(sum[k], off);
            float ov = __shfl_xor(mx[k], off);
            int   oi = __shfl_xor(si[k], off);
            if (ov > mx[k] || (ov == mx[k] && oi < si[k])) { mx[k] = ov; si[k] = oi; }
        }
    }

    // Row M = k + 8*(lane/16); one lane per row writes.
#pragma unroll
    for (int k = 0; k < 8; ++k) {
        if ((lane & 15) == k) {
            const int t = tbase + k + ((lane >> 4) << 3);
            if (t < T) {
                out_cb[t] = (float)(si[k] & 1);
                out_cw[t] = mx[k] / sum[k];
            }
        }
    }
}

// ---------------------------------------------------------------------------
// Sequential Viterbi scan: ONE wave32 owns all 256 states (8 per lane,
// s = lane*8 + r). Per step:
//   v[r]   = carry[r] + a*rx^2 + fma(-2a*p, rx, a*p^2)
//   w[p]   = min(v[2p], v[2p+1])            (intra-lane, global j = lane*4+p)
//   carry' = w[s mod 128]  -> lane reads w[(lane&15)*8 .. +7] via 2x ds_load_b128
// STORE_W: stream the 128 survivor metrics per step to d_ws (stays in 192MB L2)
// so bit extraction is a separate parallel pass; else do the argmin inline.
// ---------------------------------------------------------------------------
template <bool STORE_W>
__global__ void va_scan_kernel(const float* __restrict__ rx,
                               const float* __restrict__ h,
                               float* __restrict__ det,
                               float* __restrict__ wbuf, int T) {
    __shared__ float4 xfer[32];
    const int lane = threadIdx.x;

    float hh[8];
#pragma unroll
    for (int m = 0; m < 8; ++m) hh[m] = h[m];

    float q0[8], q1[8];
#pragma unroll
    for (int r = 0; r < 8; ++r) {
        const float p = state_prior(lane * 8 + r, hh);
        q1[r] = -2.0f * INV2SIG2 * p;
        q0[r] = INV2SIG2 * p * p;
    }

    float carry[8];
#pragma unroll
    for (int r = 0; r < 8; ++r) carry[r] = 0.0f;

    const int g2 = (lane & 15) * 2;

    for (int t0 = 0; t0 < T; t0 += 8) {
        if (t0 + 64 < T) __builtin_prefetch(rx + t0 + 64, 0, 1);
        float rb[8];
#pragma unroll
        for (int i = 0; i < 8; ++i) rb[i] = (t0 + i < T) ? rx[t0 + i] : 0.0f;

#pragma unroll
        for (int i = 0; i < 8; ++i) {
            const int t = t0 + i;
            const float rxt = rb[i];

            if (!STORE_W) {
                // bit[t] = parity of first-occurrence argmin of carry
                float lm = carry[0]; int lr = 0;
#pragma unroll
                for (int r = 1; r < 8; ++r)
                    if (carry[r] < lm) { lm = carry[r]; lr = r; }
                float gm = lm;
#pragma unroll
                for (int off = 16; off; off >>= 1) gm = fminf(gm, __shfl_xor(gm, off));
                int cand = (lm == gm) ? (lane * 8 + lr) : 0x7FFFFFFF;
#pragma unroll
                for (int off = 16; off; off >>= 1) cand = min(cand, __shfl_xor(cand, off));
                if (lane == 0 && t < T) det[t] = (float)(cand & 1);
            }

            const float k2 = INV2SIG2 * rxt * rxt;
            float v[8];
#pragma unroll
            for (int r = 0; r < 8; ++r)
                v[r] = carry[r] + (k2 + fmaf(q1[r], rxt, q0[r]));

            float4 w;
            w.x = fminf(v[0], v[1]);
            w.y = fminf(v[2], v[3]);
            w.z = fminf(v[4], v[5]);
            w.w = fminf(v[6], v[7]);

            if (STORE_W && t < T)
                ((float4*)(wbuf + (size_t)t * 128))[lane] = w;   // global_store_b128

            xfer[lane] = w;               // ds_store_b128
            __syncthreads();              // single wave -> HW NOP
            const float4 lo = xfer[g2];       // ds_load_b128
            const float4 hi = xfer[g2 + 1];   // ds_load_b128
            __syncthreads();

            carry[0] = lo.x; carry[1] = lo.y; carry[2] = lo.z; carry[3] = lo.w;
            carry[4] = hi.x; carry[5] = hi.y; carry[6] = hi.z; carry[7] = hi.w;
        }
    }
}

// ---------------------------------------------------------------------------
// Parallel bit extraction: bit[t] = parity of argmin over the 128 survivor
// metrics stored at step t-1 (in_prob duplicates halves, so argmin over 128
// with smallest-index tie-break == reference argmin over 256). bit[0] = 0.
// One wave per t; data comes back out of L2.
// ---------------------------------------------------------------------------
__global__ void va_bits_kernel(const float* __restrict__ wbuf,
                               float* __restrict__ det, int T) {
    const int wave = threadIdx.x >> 5;
    const int lane = threadIdx.x & 31;
    const int t = blockIdx.x * (blockDim.x >> 5) + wave;   // uniform per wave
    if (t >= T) return;
    if (t == 0) { if (lane == 0) det[0] = 0.0f; return; }

    const float4 wv = ((const float4*)(wbuf + (size_t)(t - 1) * 128))[lane];
    float lm = wv.x; int li = lane * 4;
    if (wv.y < lm) { lm = wv.y; li = lane * 4 + 1; }
    if (wv.z < lm) { lm = wv.z; li = lane * 4 + 2; }
    if (wv.w < lm) { lm = wv.w; li = lane * 4 + 3; }
#pragma unroll
    for (int off = 16; off; off >>= 1) {
        float ov = __shfl_xor(lm, off);
        int   oi = __shfl_xor(li, off);
        if (ov < lm || (ov == lm && oi < li)) { lm = ov; li = oi; }
    }
    if (lane == 0) det[t] = (float)(li & 1);
}

// ---------------------------------------------------------------------------
extern "C" void kernel_launch(void* const* d_in, const int* in_sizes, int n_in,
                              void* d_out, int out_size, void* d_ws, size_t ws_size,
                              hipStream_t stream) {
    const float* rx = (const float*)d_in[0];
    const float* h  = (const float*)d_in[1];
    const int T = in_sizes[0];

    float* det = (float*)d_out;              // detected_word  [T]
    float* cb  = det + (size_t)T;            // confident_bits [T]
    float* cw  = det + (size_t)2 * T;        // confidence_word[T]

    // Parallel WMMA likelihood/confidence pass: 8 waves/block, 16 t per wave.
    const int blocksC = (T + 127) / 128;
    va_conf_kernel<<<blocksC, 256, 0, stream>>>(rx, h, cb, cw, T);

    const size_t need = (size_t)T * 128 * sizeof(float);
    if (d_ws != nullptr && ws_size >= need) {
        // Short sequential chain + parallel bit extraction from L2-resident ws.
        va_scan_kernel<true><<<1, 32, 0, stream>>>(rx, h, nullptr, (float*)d_ws, T);
        va_bits_kernel<<<(T + 7) / 8, 256, 0, stream>>>((const float*)d_ws, det, T);
    } else {
        // Fallback: argmin folded into the sequential chain.
        va_scan_kernel<false><<<1, 32, 0, stream>>>(rx, h, det, nullptr, T);
    }
}